// FullFactorizedModel_68444598829586
// MI455X (gfx1250) — compile-verified
//
#include <hip/hip_runtime.h>
#include <math.h>

#define C_CH    192
#define PLANE   4096               // H*W
#define NPLANES 6144               // B*C
#define TOTAL   25165824           // NPLANES*PLANE
#define PSTRIDE 48                 // padded per-channel param stride (floats)

typedef __attribute__((ext_vector_type(4))) float v4f;   // native 128-bit vector

// ---------------------------------------------------------------------------
// Per-channel parameter layout in workspace (floats), one row per channel:
//  [0..2]   w0 = softplus(m0)        [3..5]  b0        [6..8]  t0 = tanh(f0)
//  [9..17]  W1 = softplus(m1) (row-major out,in)
//  [18..20] b1                        [21..23] t1 = tanh(f1)
//  [24..32] W2 = softplus(m2)
//  [33..35] b2                        [36..38] t2 = tanh(f2)
//  [39..41] w3 = softplus(m3)         [42] b3          [43] quantile
// ---------------------------------------------------------------------------

__global__ void ffm_setup_params(const float* __restrict__ m0, const float* __restrict__ m1,
                                 const float* __restrict__ m2, const float* __restrict__ m3,
                                 const float* __restrict__ b0, const float* __restrict__ b1,
                                 const float* __restrict__ b2, const float* __restrict__ b3,
                                 const float* __restrict__ f0, const float* __restrict__ f1,
                                 const float* __restrict__ f2, const float* __restrict__ q,
                                 float* __restrict__ ws) {
  int c = blockIdx.x * blockDim.x + threadIdx.x;
  if (c >= C_CH) return;
  float* P = ws + c * PSTRIDE;
#pragma unroll
  for (int i = 0; i < 3; ++i) {
    P[0 + i]  = log1pf(expf(m0[c * 3 + i]));   // softplus
    P[3 + i]  = b0[c * 3 + i];
    P[6 + i]  = tanhf(f0[c * 3 + i]);
    P[18 + i] = b1[c * 3 + i];
    P[21 + i] = tanhf(f1[c * 3 + i]);
    P[33 + i] = b2[c * 3 + i];
    P[36 + i] = tanhf(f2[c * 3 + i]);
    P[39 + i] = log1pf(expf(m3[c * 3 + i]));
  }
#pragma unroll
  for (int i = 0; i < 9; ++i) {
    P[9 + i]  = log1pf(expf(m1[c * 9 + i]));
    P[24 + i] = log1pf(expf(m2[c * 9 + i]));
  }
  P[42] = b3[c];
  P[43] = q[c];
  P[44] = 0.0f; P[45] = 0.0f; P[46] = 0.0f; P[47] = 0.0f;
}

// ---------------------------------------------------------------------------

struct Params {
  float w0[3], b0[3], t0[3];
  float W1[9], b1[3], t1[3];
  float W2[9], b2[3], t2[3];
  float w3[3], b3, q;
};

__device__ __forceinline__ float fast_tanh(float x) {
#if __has_builtin(__builtin_amdgcn_tanhf)
  return __builtin_amdgcn_tanhf(x);                 // v_tanh_f32 (gfx1250 HW)
#else
  // tanh(x) = 1 - 2 / (exp(2x)+1) ; exp(2x) = exp2(x * 2*log2(e))
  float e = __builtin_amdgcn_exp2f(x * 2.885390081777927f);
  return 1.0f - 2.0f * __builtin_amdgcn_rcpf(e + 1.0f);
#endif
}

// sigmoid(-t) = 1 / (1 + exp(t))
__device__ __forceinline__ float sigm_neg(float t) {
  float e = __builtin_amdgcn_exp2f(t * 1.4426950408889634f);
  return __builtin_amdgcn_rcpf(1.0f + e);
}

__device__ __forceinline__ float cdf_eval(const Params& P, float t) {
  float h[3];
#pragma unroll
  for (int i = 0; i < 3; ++i) {
    float a = fmaf(P.w0[i], t, P.b0[i]);
    h[i] = fmaf(P.t0[i], fast_tanh(a), a);
  }
  float g[3];
#pragma unroll
  for (int i = 0; i < 3; ++i) {
    float a = P.b1[i];
    a = fmaf(P.W1[i * 3 + 0], h[0], a);
    a = fmaf(P.W1[i * 3 + 1], h[1], a);
    a = fmaf(P.W1[i * 3 + 2], h[2], a);
    g[i] = fmaf(P.t1[i], fast_tanh(a), a);
  }
  float k[3];
#pragma unroll
  for (int i = 0; i < 3; ++i) {
    float a = P.b2[i];
    a = fmaf(P.W2[i * 3 + 0], g[0], a);
    a = fmaf(P.W2[i * 3 + 1], g[1], a);
    a = fmaf(P.W2[i * 3 + 2], g[2], a);
    k[i] = fmaf(P.t2[i], fast_tanh(a), a);
  }
  float o = P.b3;
  o = fmaf(P.w3[0], k[0], o);
  o = fmaf(P.w3[1], k[1], o);
  o = fmaf(P.w3[2], k[2], o);
  return o;
}

// ---------------------------------------------------------------------------
// Main kernel: one block per [b,c] plane (4096 contiguous floats).
// Streams input via async global->LDS DMA (ASYNCcnt pipeline, non-temporal),
// computes the double CDF + likelihood, stores both outputs with NT b128
// stores so the one-shot 300MB stream never sweeps the 192MB L2.
// ---------------------------------------------------------------------------
__global__ __launch_bounds__(256) void ffm_main(const float* __restrict__ x,
                                                const float* __restrict__ ws,
                                                float* __restrict__ out) {
  __shared__ float smem[PLANE];                    // 16 KB staging buffer

  const int p = blockIdx.x;                        // plane = b*C + c
  const int c = p % C_CH;
  const int t = threadIdx.x;

  const float* xplane = x + (size_t)p * PLANE;
  const unsigned lds_base = (unsigned)(size_t)(&smem[0]);

  // Issue 4 async b128 loads per thread (whole plane), tracked by ASYNCcnt.
#pragma unroll
  for (int k = 0; k < 4; ++k) {
    unsigned off  = (unsigned)((k * 256 + t) * 16);   // byte offset in plane
    unsigned ldsa = lds_base + off;
    asm volatile("global_load_async_to_lds_b128 %0, %1, %2 offset:0 th:TH_LOAD_NT"
                 :: "v"(ldsa), "v"(off), "s"(xplane) : "memory");
  }

  // Block-uniform per-channel parameters (overlaps with in-flight DMA).
  Params P;
  {
    const float* Pp = ws + c * PSTRIDE;
#pragma unroll
    for (int i = 0; i < 3; ++i) { P.w0[i] = Pp[i];      P.b0[i] = Pp[3 + i];  P.t0[i] = Pp[6 + i]; }
#pragma unroll
    for (int i = 0; i < 9; ++i) { P.W1[i] = Pp[9 + i];  P.W2[i] = Pp[24 + i]; }
#pragma unroll
    for (int i = 0; i < 3; ++i) { P.b1[i] = Pp[18 + i]; P.t1[i] = Pp[21 + i];
                                  P.b2[i] = Pp[33 + i]; P.t2[i] = Pp[36 + i];
                                  P.w3[i] = Pp[39 + i]; }
    P.b3 = Pp[42];
    P.q  = Pp[43];
  }

  float* vout = out + (size_t)p * PLANE;
  float* lout = out + (size_t)TOTAL + (size_t)p * PLANE;

  // Drain pipeline: each thread consumes exactly the LDS it staged, so a
  // per-wave ASYNCcnt wait is sufficient (loads complete in order).
#pragma unroll
  for (int k = 0; k < 4; ++k) {
    if (k == 0)      asm volatile("s_wait_asynccnt 0x3" ::: "memory");
    else if (k == 1) asm volatile("s_wait_asynccnt 0x2" ::: "memory");
    else if (k == 2) asm volatile("s_wait_asynccnt 0x1" ::: "memory");
    else             asm volatile("s_wait_asynccnt 0x0" ::: "memory");

    const int idx = (k * 256 + t) * 4;
    v4f xv = *(const v4f*)&smem[idx];               // ds_load_b128
    float xa[4] = {xv.x, xv.y, xv.z, xv.w};
    float va[4], la[4];

#pragma unroll
    for (int j = 0; j < 4; ++j) {
      float v  = rintf(xa[j] - P.q) + P.q;          // round-half-even, like jnp.round
      float lo = cdf_eval(P, v - 0.5f);
      float up = cdf_eval(P, v + 0.5f);
      float s  = ((lo + up) >= 0.0f) ? 1.0f : -1.0f; // sign with 0 -> +1
      float like = fabsf(sigm_neg(s * up) - sigm_neg(s * lo));
      va[j] = v;
      la[j] = fmaxf(like, 1e-9f);
    }

    v4f vv = {va[0], va[1], va[2], va[3]};
    v4f lv = {la[0], la[1], la[2], la[3]};
    __builtin_nontemporal_store(vv, (v4f*)&vout[idx]);   // b128 th:TH_STORE_NT
    __builtin_nontemporal_store(lv, (v4f*)&lout[idx]);   // b128 th:TH_STORE_NT
  }
}

// ---------------------------------------------------------------------------

extern "C" void kernel_launch(void* const* d_in, const int* in_sizes, int n_in,
                              void* d_out, int out_size, void* d_ws, size_t ws_size,
                              hipStream_t stream) {
  // setup_inputs order: x, m0,m1,m2,m3, b0,b1,b2,b3, f0,f1,f2, quantiles
  const float* x  = (const float*)d_in[0];
  const float* m0 = (const float*)d_in[1];
  const float* m1 = (const float*)d_in[2];
  const float* m2 = (const float*)d_in[3];
  const float* m3 = (const float*)d_in[4];
  const float* b0 = (const float*)d_in[5];
  const float* b1 = (const float*)d_in[6];
  const float* b2 = (const float*)d_in[7];
  const float* b3 = (const float*)d_in[8];
  const float* f0 = (const float*)d_in[9];
  const float* f1 = (const float*)d_in[10];
  const float* f2 = (const float*)d_in[11];
  const float* q  = (const float*)d_in[12];

  float* ws = (float*)d_ws;   // needs C_CH*PSTRIDE*4 = 36,864 bytes

  ffm_setup_params<<<1, 192, 0, stream>>>(m0, m1, m2, m3, b0, b1, b2, b3,
                                          f0, f1, f2, q, ws);
  ffm_main<<<NPLANES, 256, 0, stream>>>(x, ws, (float*)d_out);
}